// MambaEncoder_26139170964331
// MI455X (gfx1250) — compile-verified
//
#include <hip/hip_runtime.h>
#include <math.h>

// Reduced problem: output = mamba(cls)[0] broadcast over batch.
#define D_M     256
#define D_INNER 512
#define DT_RANK 16
#define XZ_COLS 1024   // 2 * D_INNER
#define WX_COLS 48     // DT_RANK + 2*D_STATE

typedef __attribute__((ext_vector_type(2))) float v2f;
typedef __attribute__((ext_vector_type(8))) float v8f;

__device__ __forceinline__ float silu_f(float v) {
    return v / (1.0f + expf(-v));
}

// ---------------------------------------------------------------------------
// K1: xz(1024) = cls(256) @ W_in(256x1024), via f32 WMMA 16x16x4 with the
// cls vector replicated across all 16 M rows. One wave per 16-column tile.
// ---------------------------------------------------------------------------
__global__ void k_xz(const float* __restrict__ cls,
                     const float* __restrict__ W_in,
                     float* __restrict__ xz_ws) {
    __shared__ float s_cls[D_M];
    const int lane = threadIdx.x & 31;
    for (int i = lane; i < D_M; i += 32) s_cls[i] = cls[i];
    __syncthreads();

    const int n0   = blockIdx.x * 16;
    const int half = lane >> 4;       // 0: lanes 0-15, 1: lanes 16-31
    const int nl   = lane & 15;
    const int ka   = half * 2;        // K offset within 4-chunk for VGPR slot

    v8f acc = {};
    for (int k = 0; k < D_M; k += 4) {
        v2f a, b;
        a.x = s_cls[k + ka];
        a.y = s_cls[k + ka + 1];
        b.x = W_in[(size_t)(k + ka)     * XZ_COLS + n0 + nl];
        b.y = W_in[(size_t)(k + ka + 1) * XZ_COLS + n0 + nl];
        acc = __builtin_amdgcn_wmma_f32_16x16x4_f32(
            false, a, false, b, (short)0, acc, false, false);
    }
    // Row M=0 lives in acc[0] of lanes 0-15.
    if (lane < 16) xz_ws[n0 + lane] = acc[0];
}

// ---------------------------------------------------------------------------
// K2: conv tap + silu, x_dbl = xc @ W_x (WMMA, 3 waves), softplus(dt@W_dt+b),
// scalar (B.C) contraction, gating; writes y(512) to workspace.
// ---------------------------------------------------------------------------
__global__ void k_mid(const float* __restrict__ xz_ws,
                      const float* __restrict__ conv_w,
                      const float* __restrict__ conv_b,
                      const float* __restrict__ W_x,
                      const float* __restrict__ W_dt,
                      const float* __restrict__ b_dt,
                      const float* __restrict__ D_skip,
                      float* __restrict__ y_ws) {
    __shared__ float s_xz[2 * D_INNER];
    __shared__ float s_xc[D_INNER];
    __shared__ float s_sz[D_INNER];
    __shared__ float s_xdbl[WX_COLS];
    __shared__ float s_bc;

    const int tid = threadIdx.x;          // 0..511
    s_xz[tid]           = xz_ws[tid];
    s_xz[tid + D_INNER] = xz_ws[tid + D_INNER];
    __syncthreads();

    {   // causal conv at t=0 keeps only the last tap; then silu. Gate silu(z).
        float v = conv_w[tid * 4 + 3] * s_xz[tid] + conv_b[tid];
        s_xc[tid] = silu_f(v);
        s_sz[tid] = silu_f(s_xz[tid + D_INNER]);
    }
    __syncthreads();

    const int wave = tid >> 5;
    const int lane = tid & 31;
    if (wave < 3) {                       // 3 tiles cover the 48 columns of W_x
        const int n0   = wave * 16;
        const int half = lane >> 4;
        const int nl   = lane & 15;
        const int ka   = half * 2;
        v8f acc = {};
        for (int k = 0; k < D_INNER; k += 4) {
            v2f a, b;
            a.x = s_xc[k + ka];
            a.y = s_xc[k + ka + 1];
            b.x = W_x[(k + ka)     * WX_COLS + n0 + nl];
            b.y = W_x[(k + ka + 1) * WX_COLS + n0 + nl];
            acc = __builtin_amdgcn_wmma_f32_16x16x4_f32(
                false, a, false, b, (short)0, acc, false, false);
        }
        if (lane < 16) s_xdbl[n0 + lane] = acc[0];
    }
    __syncthreads();

    if (tid == 0) {                       // s = Bm . Cm (scan collapses to this)
        float s = 0.0f;
        for (int n = 0; n < 16; ++n) s += s_xdbl[16 + n] * s_xdbl[32 + n];
        s_bc = s;
    }
    __syncthreads();

    {
        float acc = b_dt[tid];
        for (int r = 0; r < DT_RANK; ++r) acc += s_xdbl[r] * W_dt[r * D_INNER + tid];
        float delta = (acc > 20.0f) ? acc : log1pf(expf(acc));
        float xc = s_xc[tid];
        float y  = (delta * xc * s_bc + xc * D_skip[tid]) * s_sz[tid];
        y_ws[tid] = y;
    }
}

// ---------------------------------------------------------------------------
// K3: out(16x256) = replicate16(y(512)) @ W_out(512x256) via WMMA. The 16
// identical M rows ARE the 16 batch rows: each lane scatters its 8 C VGPRs
// to rows (half*8 + j).
// ---------------------------------------------------------------------------
__global__ void k_out(const float* __restrict__ y_ws,
                      const float* __restrict__ W_out,
                      float* __restrict__ out) {
    __shared__ float s_y[D_INNER];
    const int lane = threadIdx.x & 31;
    for (int i = lane; i < D_INNER; i += 32) s_y[i] = y_ws[i];
    __syncthreads();

    const int n0   = blockIdx.x * 16;
    const int half = lane >> 4;
    const int nl   = lane & 15;
    const int ka   = half * 2;

    v8f acc = {};
    for (int k = 0; k < D_INNER; k += 4) {
        v2f a, b;
        a.x = s_y[k + ka];
        a.y = s_y[k + ka + 1];
        b.x = W_out[(k + ka)     * D_M + n0 + nl];
        b.y = W_out[(k + ka + 1) * D_M + n0 + nl];
        acc = __builtin_amdgcn_wmma_f32_16x16x4_f32(
            false, a, false, b, (short)0, acc, false, false);
    }
    const int rbase = half * 8;           // C layout: VGPR j -> M = j (+8)
#pragma unroll
    for (int j = 0; j < 8; ++j) {
        out[(rbase + j) * D_M + n0 + nl] = acc[j];
    }
}

extern "C" void kernel_launch(void* const* d_in, const int* in_sizes, int n_in,
                              void* d_out, int out_size, void* d_ws, size_t ws_size,
                              hipStream_t stream) {
    (void)in_sizes; (void)n_in; (void)out_size; (void)ws_size;
    // setup_inputs() order:
    // 0:x 1:cls_token 2:Wp 3:bp 4:W_in 5:conv_w 6:conv_b 7:W_x 8:W_dt
    // 9:b_dt 10:A_log 11:D_skip 12:W_out
    // x/Wp/bp/A_log are provably dead for out[:,0] (causal conv + h0=0 scan).
    const float* cls    = (const float*)d_in[1];
    const float* W_in   = (const float*)d_in[4];
    const float* conv_w = (const float*)d_in[5];
    const float* conv_b = (const float*)d_in[6];
    const float* W_x    = (const float*)d_in[7];
    const float* W_dt   = (const float*)d_in[8];
    const float* b_dt   = (const float*)d_in[9];
    const float* D_skip = (const float*)d_in[11];
    const float* W_out  = (const float*)d_in[12];

    float* ws    = (float*)d_ws;
    float* xz_ws = ws;          // 1024 floats
    float* y_ws  = ws + 1024;   // 512 floats

    k_xz <<<64, 32, 0, stream>>>(cls, W_in, xz_ws);
    k_mid<<<1, 512, 0, stream>>>(xz_ws, conv_w, conv_b, W_x, W_dt, b_dt, D_skip, y_ws);
    k_out<<<16, 32, 0, stream>>>(y_ws, W_out, (float*)d_out);
}